// TorsionEmbedding_50448685859050
// MI455X (gfx1250) — compile-verified
//
#include <hip/hip_runtime.h>

#define NS 7
#define NR 6
#define NSK (NS*NR)      // 42
#define NCBF (NS*NS)     // 49
#define NOUT (NS*NS*NR)  // 294
#define TB   128         // t's per block (8 waves x 16 t)
#define NTHR 256

typedef __attribute__((ext_vector_type(4))) float v4f;
typedef __attribute__((ext_vector_type(2))) float v2f;
typedef __attribute__((ext_vector_type(8))) float v8f;
typedef __attribute__((ext_vector_type(2))) int   v2i;

static constexpr double PI_D = 3.14159265358979323846;

// ---- gfx1250 async global->LDS path (probe-verified) ----
#if defined(__gfx1250__) && defined(__has_builtin)
#  if __has_builtin(__builtin_amdgcn_global_load_async_to_lds_b64)
#    define USE_ASYNC 1
#  endif
#  if __has_builtin(__builtin_amdgcn_wmma_f32_16x16x4_f32)
#    define USE_WMMA 1
#  endif
#endif
#ifndef USE_ASYNC
#  define USE_ASYNC 0
#endif
#ifndef USE_WMMA
#  define USE_WMMA 0
#endif

#if USE_ASYNC
typedef __attribute__((address_space(1))) v2i* gv2i_p;
typedef __attribute__((address_space(3))) v2i* lv2i_p;
#endif

__device__ __forceinline__ void wait_async_zero() {
#if USE_ASYNC
#  if __has_builtin(__builtin_amdgcn_s_wait_asynccnt)
  __builtin_amdgcn_s_wait_asynccnt(0);
#  else
  asm volatile("s_wait_asynccnt 0" ::: "memory");
#  endif
#endif
}

// ---- seeds for spherical Bessel zeros (polished on device in f64) ----
__constant__ float g_zero_seed[NS][NR] = {
  { 3.14159265f,  6.28318531f,  9.42477796f, 12.56637061f, 15.70796327f, 18.84955592f},
  { 4.49340946f,  7.72525184f, 10.90412166f, 14.06619391f, 17.22075527f, 20.37130296f},
  { 5.76345920f,  9.09501133f, 12.32294097f, 15.51460301f, 18.68903636f, 21.85387422f},
  { 6.98793200f, 10.41711855f, 13.69802315f, 16.92362129f, 20.12180617f, 23.30424694f},
  { 8.18256145f, 11.70490715f, 15.03966471f, 18.30125596f, 21.52541773f, 24.72756546f},
  { 9.35581211f, 12.96653017f, 16.35470964f, 19.65315286f, 22.90455063f, 26.12775093f},
  {10.51283540f, 14.20739236f, 17.64797437f, 20.98346482f, 24.26276804f, 27.50786838f}
};

// ---- constexpr helpers ----
constexpr double cx_sqrt(double x) {
  double g = x > 1.0 ? x : 1.0;
  for (int i = 0; i < 64; ++i) g = 0.5 * (g + x / g);
  return g;
}
constexpr double cx_fact(int n) { double r = 1.0; for (int i = 2; i <= n; ++i) r *= i; return r; }

// prefactors for the direct (fallback) path
struct PrefTab { float p[NS][NS]; };
constexpr PrefTab make_pref() {
  PrefTab t{};
  for (int l = 0; l < NS; ++l)
    for (int m = 0; m < NS; ++m) t.p[l][m] = 0.0f;
  for (int l = 0; l < NS; ++l)
    for (int m = 0; m <= l; ++m) {
      double v = cx_sqrt((2.0 * l + 1.0) * cx_fact(l - m) / (4.0 * PI_D * cx_fact(l + m)));
      if (m > 0) v *= cx_sqrt(2.0);
      t.p[l][m] = (float)v;
    }
  return t;
}
__constant__ PrefTab g_pref = make_pref();

// ---- WMMA operand tables: Legendre Q_{l,m} coefficients as 8x28 B-matrix ----
// P[l][m](z) = s^m * Q_{l,m}(z);  Q evaluated as [1,z,...,z^7](K=8) x QC(8x28) matmul.
// B laid out for V_WMMA_F32_16X16X4_F32: per K-chunk c (K=4c..4c+3) and N-tile n,
// VGPR v, lane L: row K = 4c + (L>=16 ? 2:0) + v, col = n*16 + (L&15).
struct WTab {
  float B[2][2][2][32];   // [kchunk][ntile][vgpr][lane]
  float pref[28];         // sqrt2-included prefactor per (l,m) pair
  int   lm[28];           // l*8+m
};
constexpr WTab make_wtab() {
  WTab t{};
  double qc[7][7][8] = {};
  for (int m = 0; m < 7; ++m) {
    double dm = 1.0;
    for (int k = 1; k <= m; ++k) dm *= (double)(1 - 2 * k);
    qc[m][m][0] = dm;                                  // Q[m][m] = prod(1-2k)
    if (m + 1 < 7)
      for (int p = 0; p < 7; ++p) qc[m + 1][m][p + 1] = (double)(2 * m + 1) * qc[m][m][p];
    for (int l = m + 2; l < 7; ++l)
      for (int p = 0; p < 8; ++p) {
        double a = (p > 0) ? qc[l - 1][m][p - 1] : 0.0;
        qc[l][m][p] = ((double)(2 * l - 1) * a - (double)(l + m - 1) * qc[l - 2][m][p])
                      / (double)(l - m);
      }
  }
  int l28[28] = {}, m28[28] = {};
  {
    int q = 0;
    for (int l = 0; l < 7; ++l)
      for (int m = 0; m <= l; ++m) { l28[q] = l; m28[q] = m; ++q; }
  }
  for (int q = 0; q < 28; ++q) {
    int l = l28[q], m = m28[q];
    double pv = cx_sqrt((2.0 * l + 1.0) * cx_fact(l - m) / (4.0 * PI_D * cx_fact(l + m)));
    if (m > 0) pv *= cx_sqrt(2.0);
    t.pref[q] = (float)pv;
    t.lm[q] = l * 8 + m;
  }
  for (int c = 0; c < 2; ++c)
    for (int n = 0; n < 2; ++n)
      for (int vv = 0; vv < 2; ++vv)
        for (int lane = 0; lane < 32; ++lane) {
          int K   = 4 * c + ((lane >= 16) ? 2 : 0) + vv;
          int col = n * 16 + (lane & 15);
          float val = 0.0f;
          if (col < 28) val = (float)qc[l28[col]][m28[col]][K];
          t.B[c][n][vv][lane] = val;
        }
  return t;
}
__constant__ WTab g_w = make_wtab();

// ---- f64 spherical Bessel j_n (upward recurrence; x > n here, stable) ----
__device__ double sph_jn_d(double x, int n) {
  double s = sin(x), c = cos(x);
  double jm = s / x;
  if (n == 0) return jm;
  double jc = s / (x * x) - c / x;
  for (int l = 2; l <= n; ++l) { double t = (2.0 * l - 1.0) / x * jc - jm; jm = jc; jc = t; }
  return jc;
}

// Kernel 1: polish zeros (Newton, f64) + norms -> workspace
__global__ void init_kernel(float* __restrict__ zeros_out, float* __restrict__ norms_out) {
  int i = threadIdx.x;
  if (i >= NSK) return;
  int l = i / NR, j = i - l * NR;
  double x;
  if (l == 0) {
    x = PI_D * (double)(j + 1);
  } else {
    x = (double)g_zero_seed[l][j];
    for (int it = 0; it < 8; ++it) {
      double f  = sph_jn_d(x, l);
      double fm = sph_jn_d(x, l - 1);
      double fp = fm - (double)(l + 1) / x * f;   // j_l'(x)
      x -= f / fp;
    }
  }
  double jn1 = sph_jn_d(x, l + 1);
  zeros_out[i] = (float)x;
  norms_out[i] = (float)(1.0 / sqrt(0.5 * jn1 * jn1));
}

// Kernel 2: rbf table, E x 42 (one row per edge); 25 MB -> stays L2-resident
__global__ __launch_bounds__(256) void rbf_table_kernel(
    const float* __restrict__ dist, const float* __restrict__ zeros,
    const float* __restrict__ norms, float* __restrict__ table, int E)
{
  int e = blockIdx.x * blockDim.x + threadIdx.x;
  if (e >= E) return;
  const float dr = dist[e] * 0.2f;   // dist / CUTOFF
  float o[NSK];
#pragma unroll
  for (int l = 0; l < NS; ++l) {
#pragma unroll
    for (int j = 0; j < NR; ++j) {
      const float x = dr * zeros[l * NR + j];
      float sx, cx;
      sincosf(x, &sx, &cx);
      const float inv = 1.0f / x;
      float jm = sx * inv;
      float jc = (l == 0) ? jm : (sx * inv - cx) * inv;
#pragma unroll
      for (int ll = 2; ll <= l; ++ll) {
        float t = (float)(2 * ll - 1) * inv * jc - jm;
        jm = jc; jc = t;
      }
      o[l * NR + j] = norms[l * NR + j] * jc;
    }
  }
  float* dst = table + (size_t)e * NSK;
#pragma unroll
  for (int i = 0; i < NSK; ++i) dst[i] = o[i];
}

// Kernel 3: async-gather rbf rows to LDS, cbf via WMMA Legendre matmul, stream out
__global__ __launch_bounds__(NTHR) void torsion_expand_kernel(
    const float* __restrict__ angle, const float* __restrict__ phi,
    const int* __restrict__ idx_kj, const float* __restrict__ rbf_table,
    float* __restrict__ out, int T)
{
  __shared__ float sCbf[TB * NCBF];
  __shared__ float sRbf[TB * NSK];
  __shared__ float sG[TB * 13];     // per-t: [m]=cos(m phi)*s^m (m=0..6), [6+m]=sin(m phi)*s^m (m=1..6)
  __shared__ int   sIdx[TB];

  const int tid  = threadIdx.x;
  const int base = blockIdx.x * TB;
  int rem = T - base;
  const int nt = rem < TB ? rem : TB;

  if (tid < nt) sIdx[tid] = idx_kj[base + tid];
  __syncthreads();

  // Gather 42-f32 rbf rows as b64 chunks (rows are 8B-aligned: 42*4=168B).
  const int npairs = nt * (NSK / 2);   // 21 dword-pairs per row
  for (int c = tid; c < npairs; c += NTHR) {
    int tl = c / 21;
    int h  = c - tl * 21;
    const float* src = rbf_table + (size_t)sIdx[tl] * NSK + h * 2;
    float* dst = &sRbf[tl * NSK + h * 2];
#if USE_ASYNC
    __builtin_amdgcn_global_load_async_to_lds_b64((gv2i_p)src, (lv2i_p)dst, 0, 0);
#else
    dst[0] = src[0];
    dst[1] = src[1];
#endif
  }

#if USE_WMMA
  {
    const int lane = tid & 31;
    const int w    = tid >> 5;          // wave id: handles t-block [w*16, w*16+16)
    const int half = lane >> 4;
    const int l15  = lane & 15;

    // A operand: monomials of z = cos(angle_t), 16 t per wave, K=8 in 2 chunks.
    int tA = base + w * 16 + l15; if (tA > T - 1) tA = T - 1;
    const float zA = cosf(angle[tA]);
    const float z2 = zA * zA, z3 = z2 * zA, z4 = z2 * z2;
    const float z5 = z4 * zA, z6 = z4 * z2, z7 = z4 * z3;
    v2f a0, a1;
    a0.x = half ? z2 : 1.0f;  a0.y = half ? z3 : zA;
    a1.x = half ? z6 : z4;    a1.y = half ? z7 : z5;

    v2f b00, b10, b01, b11;
    b00.x = g_w.B[0][0][0][lane]; b00.y = g_w.B[0][0][1][lane];
    b10.x = g_w.B[1][0][0][lane]; b10.y = g_w.B[1][0][1][lane];
    b01.x = g_w.B[0][1][0][lane]; b01.y = g_w.B[0][1][1][lane];
    b11.x = g_w.B[1][1][0][lane]; b11.y = g_w.B[1][1][1][lane];

    v8f acc0 = {0.f,0.f,0.f,0.f,0.f,0.f,0.f,0.f};
    v8f acc1 = {0.f,0.f,0.f,0.f,0.f,0.f,0.f,0.f};
    // Q(16t x 28) = monomials(16x8) x QC(8x28): 4 x V_WMMA_F32_16X16X4_F32
    acc0 = __builtin_amdgcn_wmma_f32_16x16x4_f32(false, a0, false, b00, (short)0, acc0, false, false);
    acc0 = __builtin_amdgcn_wmma_f32_16x16x4_f32(false, a1, false, b10, (short)0, acc0, false, false);
    acc1 = __builtin_amdgcn_wmma_f32_16x16x4_f32(false, a0, false, b01, (short)0, acc1, false, false);
    acc1 = __builtin_amdgcn_wmma_f32_16x16x4_f32(false, a1, false, b11, (short)0, acc1, false, false);

    // Per-t trig * s^m factors -> LDS (lanes 0-15 of each wave, one t each)
    if (!half) {
      const float ang = angle[tA];   // tA == clamped(base + w*16 + l15)
      int tP = base + w * 16 + l15; if (tP > T - 1) tP = T - 1;
      const float ph = phi[tP];
      const float s  = sinf(ang);
      float sp, cp;
      sincosf(ph, &sp, &cp);
      const int gb = (w * 16 + l15) * 13;
      sG[gb + 0] = 1.0f;
      float cmm = cp, smm = sp, cm1 = 1.0f, sm1 = 0.0f;
      const float twoc = 2.0f * cp;
      float pw = s;
#pragma unroll
      for (int m = 1; m < NS; ++m) {
        sG[gb + m]     = cmm * pw;
        sG[gb + 6 + m] = smm * pw;
        float cn = twoc * cmm - cm1, sn = twoc * smm - sm1;
        cm1 = cmm; sm1 = smm; cmm = cn; smm = sn;
        pw *= s;
      }
    }
    __syncthreads();   // make sG visible to lanes 16-31

    // Combine: cbf[l^2+l+m] = pref*Q*G[m], cbf[l^2+l-m] = pref*Q*G[6+m]
#pragma unroll
    for (int n = 0; n < 2; ++n) {
      const v8f acc = n ? acc1 : acc0;
      const int col = n * 16 + l15;
      if (col < 28) {
        const int lmv = g_w.lm[col];
        const int ll = lmv >> 3, mm = lmv & 7;
        const float pf = g_w.pref[col];
#pragma unroll
        for (int r = 0; r < 8; ++r) {
          const int tb = w * 16 + (half ? 8 + r : r);
          if (tb < nt) {
            const float vq = pf * acc[r];
            const int cb = tb * NCBF + ll * ll + ll;
            sCbf[cb + mm] = vq * sG[tb * 13 + mm];
            if (mm) sCbf[cb - mm] = vq * sG[tb * 13 + 6 + mm];
          }
        }
      }
    }
  }
#else
  // Fallback: direct recurrences, one t per thread
  if (tid < nt) {
    const float ang = angle[base + tid];
    const float ph  = phi[base + tid];
    float s = sinf(ang);
    float z = cosf(ang);
    float sp, cp;
    sincosf(ph, &sp, &cp);

    float P[NS][NS];
    P[0][0] = 1.0f;
#pragma unroll
    for (int m = 1; m < NS; ++m) P[m][m] = (float)(1 - 2 * m) * s * P[m - 1][m - 1];
#pragma unroll
    for (int m = 0; m < NS - 1; ++m) P[m + 1][m] = (float)(2 * m + 1) * z * P[m][m];
#pragma unroll
    for (int m = 0; m < NS; ++m) {
#pragma unroll
      for (int l = m + 2; l < NS; ++l) {
        P[l][m] = ((float)(2 * l - 1) * z * P[l - 1][m]
                   - (float)(l + m - 1) * P[l - 2][m]) * (1.0f / (float)(l - m));
      }
    }
    float cm[NS], sm[NS];
    cm[0] = 1.0f; sm[0] = 0.0f;
    cm[1] = cp;   sm[1] = sp;
    const float c2 = 2.0f * cp;
#pragma unroll
    for (int m = 2; m < NS; ++m) {
      cm[m] = c2 * cm[m - 1] - cm[m - 2];
      sm[m] = c2 * sm[m - 1] - sm[m - 2];
    }
    float* dst = &sCbf[tid * NCBF];
    dst[0] = g_pref.p[0][0];
    int col = 1;
#pragma unroll
    for (int l = 1; l < NS; ++l) {
#pragma unroll
      for (int m = -6; m <= 6; ++m) {
        if (m < -l || m > l) continue;
        int am = m < 0 ? -m : m;
        float v = g_pref.p[l][am] * P[l][am];
        v *= (m < 0) ? sm[am] : cm[am];
        dst[col++] = v;
      }
    }
  }
#endif

  wait_async_zero();
  __syncthreads();

  // Stream output: 294 contiguous f32 per t; block chunk is float4-aligned.
  const size_t outBase = (size_t)base * NOUT;
  const int total  = nt * NOUT;
  const int total4 = total >> 2;
  v4f* out4 = (v4f*)(out + outBase);
  for (int q = tid; q < total4; q += NTHR) {
    int e = q << 2;
    v4f v;
#pragma unroll
    for (int u = 0; u < 4; ++u) {
      int ee = e + u;
      int tl = ee / NOUT;
      int i  = ee - tl * NOUT;
      v[u] = sCbf[tl * NCBF + i / NR] * sRbf[tl * NSK + (i % NSK)];
    }
    __builtin_nontemporal_store(v, &out4[q]);   // NT: keep L2 for the rbf table
  }
  for (int ee = (total4 << 2) + tid; ee < total; ee += NTHR) {
    int tl = ee / NOUT;
    int i  = ee - tl * NOUT;
    out[outBase + ee] = sCbf[tl * NCBF + i / NR] * sRbf[tl * NSK + (i % NSK)];
  }
}

extern "C" void kernel_launch(void* const* d_in, const int* in_sizes, int n_in,
                              void* d_out, int out_size, void* d_ws, size_t ws_size,
                              hipStream_t stream) {
  const float* dist  = (const float*)d_in[0];
  const float* angle = (const float*)d_in[1];
  const float* phi   = (const float*)d_in[2];
  const int*   idx   = (const int*)d_in[3];
  float* out = (float*)d_out;
  const int E = in_sizes[0];
  const int T = in_sizes[1];

  float* wsf   = (float*)d_ws;
  float* zeros = wsf;          // [0..41]
  float* norms = wsf + 64;     // [64..105]
  float* table = wsf + 128;    // E*42 floats (~25 MB)

  init_kernel<<<1, 64, 0, stream>>>(zeros, norms);
  rbf_table_kernel<<<(E + 255) / 256, 256, 0, stream>>>(dist, zeros, norms, table, E);
  torsion_expand_kernel<<<(T + TB - 1) / TB, NTHR, 0, stream>>>(angle, phi, idx, table, out, T);
}